// DTWLoss_67422396612860
// MI455X (gfx1250) — compile-verified
//
#include <hip/hip_runtime.h>
#include <math.h>

#define L 512
#define NB 64
// gamma = 0.1, folded into base-2 exp/log:
//   exp(-(r-m)/g) = exp2((m-r) * KEXP),  KEXP = log2(e)/gamma
//   g * ln(e_sum) = KLOG * log2(e_sum),  KLOG = gamma * ln(2)
#define KEXP 14.426950408889634f
#define KLOG 0.06931471805599453f

// One workgroup per batch element. 512 threads = 16 wave32 waves sweep the
// anti-diagonals of the (L+1)x(L+1) soft-DTW DP table; thread t owns DP
// column t+1. Register carries: r1 (own previous diag value) and r0 (left
// neighbor's value from two diags ago == previous step's r2 load), so each
// step does just one DP load + one DP store; two ping-pong LDS buffers.
__global__ __launch_bounds__(512) void sdtw_diag_kernel(
    const float* __restrict__ pred,
    const float* __restrict__ target,
    float* __restrict__ per_batch) {
  __shared__ float sP0[L + 1];
  __shared__ float sP1[L + 1];
  __shared__ float s_pred[L];

  const int tid = threadIdx.x;      // 0..511
  const int b   = blockIdx.x;       // batch index
  const int col = tid + 1;          // DP column 1..512
  const float INF = __builtin_inff();

  // --- CDNA5 async global->LDS DMA: stage this batch's pred row into LDS.
  // Each lane copies one dword; tracked on ASYNCcnt (gfx1250-specific path).
  {
    unsigned int lds_addr =
        (unsigned int)(unsigned long long)(&s_pred[tid]);   // low 32b = LDS offset
    unsigned long long gaddr =
        (unsigned long long)(pred + (size_t)b * L + tid);
    asm volatile("global_load_async_to_lds_b32 %0, %1, off"
                 :: "v"(lds_addr), "v"(gaddr)
                 : "memory");
  }

  // Target value for this column stays in a register for the whole sweep.
  const float tv = target[(size_t)b * L + tid];

  // Only slot 0 (the R[row][0] = +inf boundary) ever needs initialization;
  // every other slot is written by its owning thread before first read.
  if (tid == 0) { sP0[0] = INF; sP1[0] = INF; }

  asm volatile("s_wait_asynccnt 0" ::: "memory");   // pred row resident in LDS
  __syncthreads();

  // Register carries for the upcoming step:
  float r0c = INF;   // R[i-1][col-1]  (diag d-2, left neighbor)
  float r1c = INF;   // R[i-1][col]    (own value from diag d-1)

  // Peeled d=2: only col==1 active; R[1][1] = (p0-t0)^2 + softmin(0,inf,inf)
  // and softmin(0,inf,inf) == 0 exactly.
  {
    if (col == 1) {
      const float diff = s_pred[0] - tv;
      const float val = diff * diff;
      sP0[col] = val;
      r1c = val;
    }
    // r0c stays INF: next step's R[i-1][col-1] is a diag-1 boundary (inf).
    __syncthreads();
  }

  // One anti-diagonal step. All threads reach the barrier uniformly; the
  // guarded body only runs for cells with 1 <= i <= L.
  auto step = [&](const float* __restrict__ prev, float* __restrict__ cur,
                  int i) {
    if ((unsigned)(i - 1) < (unsigned)L) {
      const float r2 = prev[col - 1];            // R[i][col-1] (diag d-1, left)
      const float m = fminf(fminf(r0c, r1c), r2);
      float val = INF;
      if (m < 3.0e38f) {                         // finite: stable softmin
        const float mK = m * KEXP;
        const float e = exp2f(fmaf(r0c, -KEXP, mK)) +
                        exp2f(fmaf(r1c, -KEXP, mK)) +
                        exp2f(fmaf(r2,  -KEXP, mK));
        const float diff = s_pred[i - 1] - tv;
        val = fmaf(diff, diff, fmaf(-KLOG, __log2f(e), m));
      }
      cur[col] = val;
      r0c = r2;                                  // becomes R[i-1][col-1] next step
      r1c = val;                                 // becomes R[i-1][col]   next step
    }
    __syncthreads();
  };

  // d = 3 .. 1024: 1022 steps = 511 ping-pong pairs.
  int i = 3 - col;
#pragma unroll 1
  for (int k = 0; k < 511; ++k) {
    step(sP0, sP1, i); ++i;
    step(sP1, sP0, i); ++i;
  }

  // After d = 2L, thread col==L holds R[L][L] in its r1 carry.
  if (col == L) per_batch[b] = r1c;
}

// Epilogue: mean of the 64 per-batch losses via a single wave32 + shuffles.
__global__ void sdtw_mean_kernel(const float* __restrict__ per_batch,
                                 float* __restrict__ out) {
  const int lane = threadIdx.x;    // 0..31
  float s = per_batch[lane] + per_batch[lane + 32];
  for (int off = 16; off > 0; off >>= 1)
    s += __shfl_down(s, off, 32);
  if (lane == 0) out[0] = s * (1.0f / (float)NB);
}

extern "C" void kernel_launch(void* const* d_in, const int* in_sizes, int n_in,
                              void* d_out, int out_size, void* d_ws, size_t ws_size,
                              hipStream_t stream) {
  const float* pred   = (const float*)d_in[0];   // [64, 512] fp32
  const float* target = (const float*)d_in[1];   // [64, 512] fp32
  float* per_batch = (float*)d_ws;               // 64 floats of scratch

  sdtw_diag_kernel<<<NB, L, 0, stream>>>(pred, target, per_batch);
  sdtw_mean_kernel<<<1, 32, 0, stream>>>(per_batch, (float*)d_out);
}